// GATv2_31988916421123
// MI455X (gfx1250) — compile-verified
//
#include <hip/hip_runtime.h>
#include <math.h>

#define N_NODES 50000
#define N_EDGES 800000
#define DH 64
#define NG 64
#define DOUT 16
#define NEG_SLOPE 0.2f

typedef __attribute__((ext_vector_type(2))) float v2f;
typedef __attribute__((ext_vector_type(8))) float v8f;

// ---------------------------------------------------------------------------
// Node transform: XL = H*Wl + bl, XR = H*Wr + br.  H:[N,64], W:[64,64].
// One block = one 16-node tile; 4 waves, each wave owns 16 output columns and
// computes BOTH the Wl and Wr tiles (A fragment reused across the two WMMAs).
// fp32 WMMA 16x16x4: A lane layout r=lane&15, kb=2*(lane>>4);
// C/D layout: VGPR i -> M = i + 8*(lane>>4), N = lane&15.
// ---------------------------------------------------------------------------
#if __has_builtin(__builtin_amdgcn_wmma_f32_16x16x4_f32)
__global__ void k_node_transform(const float* __restrict__ H,
                                 const float* __restrict__ Wl, const float* __restrict__ bl,
                                 const float* __restrict__ Wr, const float* __restrict__ br,
                                 float* __restrict__ XL, float* __restrict__ XR) {
  const int lane  = threadIdx.x & 31;
  const int ntile = threadIdx.x >> 5;          // 0..3 -> 16-column slice
  const int r     = lane & 15;
  const int hi    = lane >> 4;                 // 0 or 1
  const int kb    = hi << 1;                   // 0 or 2
  const int node0 = blockIdx.x << 4;
  const float* hrow = H + (size_t)(node0 + r) * DH;
  const int col = (ntile << 4) + r;

  v8f cl = {};
  v8f cr = {};
#pragma unroll
  for (int k = 0; k < DH; k += 4) {
    v2f a, bL, bR;
    a[0]  = hrow[k + kb];
    a[1]  = hrow[k + kb + 1];
    bL[0] = Wl[(k + kb) * DH + col];
    bL[1] = Wl[(k + kb + 1) * DH + col];
    bR[0] = Wr[(k + kb) * DH + col];
    bR[1] = Wr[(k + kb + 1) * DH + col];
    cl = __builtin_amdgcn_wmma_f32_16x16x4_f32(false, a, false, bL, (short)0, cl, false, false);
    cr = __builtin_amdgcn_wmma_f32_16x16x4_f32(false, a, false, bR, (short)0, cr, false, false);
  }

  const int   mbase = node0 + (hi << 3);
  const float bll = bl[col], brr = br[col];
#pragma unroll
  for (int i = 0; i < 8; ++i) {
    XL[(size_t)(mbase + i) * DH + col] = cl[i] + bll;
    XR[(size_t)(mbase + i) * DH + col] = cr[i] + brr;
  }
}
#else
// VALU fallback (same launch geometry) — should not be taken on gfx1250.
__global__ void k_node_transform(const float* __restrict__ H,
                                 const float* __restrict__ Wl, const float* __restrict__ bl,
                                 const float* __restrict__ Wr, const float* __restrict__ br,
                                 float* __restrict__ XL, float* __restrict__ XR) {
  const int node0 = blockIdx.x << 4;
  for (int t = threadIdx.x; t < 16 * DH; t += blockDim.x) {
    const int rr = t >> 6, col = t & 63;
    const float* hrow = H + (size_t)(node0 + rr) * DH;
    float al = bl[col], ar = br[col];
    for (int k = 0; k < DH; ++k) {
      const float hv = hrow[k];
      al += hv * Wl[k * DH + col];
      ar += hv * Wr[k * DH + col];
    }
    XL[(size_t)(node0 + rr) * DH + col] = al;
    XR[(size_t)(node0 + rr) * DH + col] = ar;
  }
}
#endif

// ---------------------------------------------------------------------------
// Per-layer scratch init: acc = 0, m = -inf, s = 0
// ---------------------------------------------------------------------------
__global__ void k_init(float* __restrict__ acc, float* __restrict__ mbuf,
                       float* __restrict__ sbuf) {
  const int i = blockIdx.x * blockDim.x + threadIdx.x;
  if (i < N_NODES * DH) acc[i] = 0.f;
  if (i < N_NODES) { mbuf[i] = -__builtin_inff(); sbuf[i] = 0.f; }
}

// ---------------------------------------------------------------------------
// Per-edge logits + segment-max.  Fully fused: e = lrelu(xl[s]+xr[d]+ea*We),
// logit = e.att.  float4 (B128) loads; gathers hit L2 (12.8MB tables).
// Float atomic-max via the int/uint-punning trick.
// ---------------------------------------------------------------------------
__global__ void k_edge_logits(const float* __restrict__ XL, const float* __restrict__ XR,
                              const float* __restrict__ ea,
                              const int* __restrict__ src, const int* __restrict__ dst,
                              const float* __restrict__ We, const float* __restrict__ att,
                              float* __restrict__ logit, float* __restrict__ mbuf) {
  const int e = blockIdx.x * blockDim.x + threadIdx.x;
  if (e >= N_EDGES) return;
  const int s = src[e], d = dst[e];
  const float w = ea[e];
  const float4* pl = (const float4*)(XL + (size_t)s * DH);
  const float4* pr = (const float4*)(XR + (size_t)d * DH);
  const float4* pw = (const float4*)We;
  const float4* pa = (const float4*)att;
  float acc = 0.f;
#pragma unroll
  for (int i = 0; i < DH / 4; ++i) {
    const float4 a = pl[i], b = pr[i], ww = pw[i], at = pa[i];
    float v0 = a.x + b.x + w * ww.x; v0 = v0 > 0.f ? v0 : v0 * NEG_SLOPE;
    float v1 = a.y + b.y + w * ww.y; v1 = v1 > 0.f ? v1 : v1 * NEG_SLOPE;
    float v2 = a.z + b.z + w * ww.z; v2 = v2 > 0.f ? v2 : v2 * NEG_SLOPE;
    float v3 = a.w + b.w + w * ww.w; v3 = v3 > 0.f ? v3 : v3 * NEG_SLOPE;
    acc += at.x * v0 + at.y * v1 + at.z * v2 + at.w * v3;
  }
  logit[e] = acc;
  if (acc >= 0.f) atomicMax((int*)(mbuf + d), __float_as_int(acc));
  else            atomicMin((unsigned int*)(mbuf + d), (unsigned int)__float_as_int(acc));
}

// p = exp(logit - m[dst]); segment-sum p into s (L2 float atomics)
__global__ void k_edge_exp(float* __restrict__ logit, const float* __restrict__ mbuf,
                           const int* __restrict__ dst, float* __restrict__ sbuf) {
  const int e = blockIdx.x * blockDim.x + threadIdx.x;
  if (e >= N_EDGES) return;
  const int d = dst[e];
  const float p = expf(logit[e] - mbuf[d]);
  logit[e] = p;
  atomicAdd(sbuf + d, p);
}

// acc[dst] += alpha * xl[src];  16 threads/edge, each owns one float4 chunk
__global__ void k_edge_scatter(const float* __restrict__ p, const float* __restrict__ sbuf,
                               const float* __restrict__ XL,
                               const int* __restrict__ src, const int* __restrict__ dst,
                               float* __restrict__ acc) {
  const int idx = blockIdx.x * blockDim.x + threadIdx.x;
  const int e = idx >> 4;
  if (e >= N_EDGES) return;
  const int q = idx & 15;
  const int d = dst[e];
  const float alpha = p[e] / sbuf[d];
  const float4 v = ((const float4*)(XL + (size_t)src[e] * DH))[q];
  float* o = acc + (size_t)d * DH + (q << 2);
  atomicAdd(o + 0, alpha * v.x);
  atomicAdd(o + 1, alpha * v.y);
  atomicAdd(o + 2, alpha * v.z);
  atomicAdd(o + 3, alpha * v.w);
}

// h = relu(acc + b)  (layers 1,2)
__global__ void k_finalize_relu(const float* __restrict__ acc, const float* __restrict__ b,
                                float* __restrict__ h) {
  const int i = blockIdx.x * blockDim.x + threadIdx.x;
  if (i >= N_NODES * DH) return;
  h[i] = fmaxf(acc[i] + b[i & 63], 0.f);
}

__global__ void k_pool_init(float* __restrict__ pool, float* __restrict__ cnt) {
  const int i = blockIdx.x * blockDim.x + threadIdx.x;
  if (i < NG * DH) pool[i] = 0.f;
  if (i < NG) cnt[i] = 0.f;
}

// layer-3 finalize fused with mean-pool accumulation
__global__ void k_finalize_pool(const float* __restrict__ acc, const float* __restrict__ b,
                                const int* __restrict__ batch,
                                float* __restrict__ pool, float* __restrict__ cnt) {
  const int i = blockIdx.x * blockDim.x + threadIdx.x;
  if (i >= N_NODES * DH) return;
  const int n = i >> 6, d = i & 63;
  const int g = batch[n];
  atomicAdd(pool + g * DH + d, acc[i] + b[d]);
  if (d == 0) atomicAdd(cnt + g, 1.0f);
}

// out[g,o] = blin[o] + (pool[g,:]/max(cnt,1)) . Wlin[:,o]   (tiny)
__global__ void k_head(const float* __restrict__ pool, const float* __restrict__ cnt,
                       const float* __restrict__ Wlin, const float* __restrict__ blin,
                       float* __restrict__ out) {
  const int t = blockIdx.x * blockDim.x + threadIdx.x;
  if (t >= NG * DOUT) return;
  const int g = t >> 4, o = t & 15;
  const float inv = 1.0f / fmaxf(cnt[g], 1.0f);
  float a = blin[o];
#pragma unroll
  for (int k = 0; k < DH; ++k) a += pool[g * DH + k] * inv * Wlin[k * DOUT + o];
  out[t] = a;
}

// ---------------------------------------------------------------------------
extern "C" void kernel_launch(void* const* d_in, const int* in_sizes, int n_in,
                              void* d_out, int out_size, void* d_ws, size_t ws_size,
                              hipStream_t stream) {
  const float* x     = (const float*)d_in[0];
  const int*   ei    = (const int*)d_in[1];
  const int*   src   = ei;
  const int*   dstp  = ei + N_EDGES;
  const float* ea    = (const float*)d_in[2];
  const int*   batch = (const int*)d_in[3];

  const float* Wl[3]  = {(const float*)d_in[4],  (const float*)d_in[11], (const float*)d_in[18]};
  const float* bl[3]  = {(const float*)d_in[5],  (const float*)d_in[12], (const float*)d_in[19]};
  const float* Wr[3]  = {(const float*)d_in[6],  (const float*)d_in[13], (const float*)d_in[20]};
  const float* br[3]  = {(const float*)d_in[7],  (const float*)d_in[14], (const float*)d_in[21]};
  const float* We[3]  = {(const float*)d_in[8],  (const float*)d_in[15], (const float*)d_in[22]};
  const float* att[3] = {(const float*)d_in[9],  (const float*)d_in[16], (const float*)d_in[23]};
  const float* bo[3]  = {(const float*)d_in[10], (const float*)d_in[17], (const float*)d_in[24]};
  const float* Wlin = (const float*)d_in[25];
  const float* blin = (const float*)d_in[26];

  float* ws   = (float*)d_ws;
  float* hbuf = ws;                         // N*64
  float* xl   = hbuf + (size_t)N_NODES * DH;
  float* xr   = xl   + (size_t)N_NODES * DH;
  float* accb = xr   + (size_t)N_NODES * DH;
  float* pbuf = accb + (size_t)N_NODES * DH; // E (logits, then p)
  float* mbuf = pbuf + N_EDGES;              // N
  float* sbuf = mbuf + N_NODES;              // N
  float* pool = sbuf + N_NODES;              // G*64
  float* cnt  = pool + NG * DH;              // G
  float* out  = (float*)d_out;

  const int gND  = (N_NODES * DH + 255) / 256;
  const int gE   = (N_EDGES + 255) / 256;
  const int gE16 = (N_EDGES * 16 + 255) / 256;
  const int gMM  = N_NODES / 16;            // 50000/16 = 3125 exactly

  for (int li = 0; li < 3; ++li) {
    const float* hin = (li == 0) ? x : hbuf;
    k_init<<<gND, 256, 0, stream>>>(accb, mbuf, sbuf);
    k_node_transform<<<gMM, 128, 0, stream>>>(hin, Wl[li], bl[li], Wr[li], br[li], xl, xr);
    k_edge_logits<<<gE, 256, 0, stream>>>(xl, xr, ea, src, dstp, We[li], att[li], pbuf, mbuf);
    k_edge_exp<<<gE, 256, 0, stream>>>(pbuf, mbuf, dstp, sbuf);
    k_edge_scatter<<<gE16, 256, 0, stream>>>(pbuf, sbuf, xl, src, dstp, accb);
    if (li < 2) k_finalize_relu<<<gND, 256, 0, stream>>>(accb, bo[li], hbuf);
  }
  k_pool_init<<<(NG * DH + 255) / 256, 256, 0, stream>>>(pool, cnt);
  k_finalize_pool<<<gND, 256, 0, stream>>>(accb, bo[2], batch, pool, cnt);
  k_head<<<(NG * DOUT + 255) / 256, 256, 0, stream>>>(pool, cnt, Wlin, blin, out);
}